// TargetBroadcastTC_35777077575988
// MI455X (gfx1250) — compile-verified
//
#include <hip/hip_runtime.h>
#include <stdint.h>

#define PP       8
#define BCN      (1024 * 1024)   // B*C flattened batch
#define TILE     64              // bc pairs per workgroup
#define THREADS  256
#define NWAVES   8
#define SROW     17              // padded scratch row stride (floats) -> no bank conflicts

typedef __attribute__((ext_vector_type(2))) float v2f;
typedef __attribute__((ext_vector_type(8))) float v8f;

// Low 32 bits of a flat shared pointer are the wave-relative LDS byte offset.
__device__ __forceinline__ uint32_t lds_off(const void* p) {
    return (uint32_t)(uintptr_t)p;
}

__device__ __forceinline__ void async_load_b128(uint32_t dst_lds, const float* gsrc) {
    asm volatile("global_load_async_to_lds_b128 %0, %1, off"
                 :: "v"(dst_lds), "v"((uint64_t)(uintptr_t)gsrc)
                 : "memory");
}
__device__ __forceinline__ void async_store_b128(float* gdst, uint32_t src_lds) {
    asm volatile("global_store_async_from_lds_b128 %0, %1, off"
                 :: "v"((uint64_t)(uintptr_t)gdst), "v"(src_lds)
                 : "memory");
}
__device__ __forceinline__ void wait_async0() {
    asm volatile("s_wait_asynccnt 0" ::: "memory");
}

__device__ __forceinline__ v8f wmma_f32_4(v2f a, v2f b, v8f c) {
    return __builtin_amdgcn_wmma_f32_16x16x4_f32(false, a, false, b,
                                                 (short)0, c, false, false);
}

__global__ __launch_bounds__(THREADS)
void tc_bcast_wmma_kernel(const float* __restrict__ T1,
                          const float* __restrict__ T2,
                          float* __restrict__ out) {
    __shared__ float sT1[PP][TILE * 9];       // 18432 B
    __shared__ float sT2[TILE * 9];           //  2304 B
    __shared__ float sOut[PP][TILE * 9];      // 18432 B
    __shared__ float sD[NWAVES * 16 * SROW];  //  8704 B  per-wave D scratch
    __shared__ float sDummy[4];               // sink for invalid-lane stores

    const int    tid    = threadIdx.x;
    const size_t bcBase = (size_t)blockIdx.x * TILE;

    // ---- stage tile into LDS: T1 = 8 segments x 144 float4, T2 = 144 float4
    for (int c = tid; c < 1296; c += THREADS) {
        if (c < 1152) {
            const int p = c / 144, rem = c % 144;
            const float* g = T1 + ((size_t)p * BCN + bcBase) * 9 + rem * 4;
            async_load_b128(lds_off(&sT1[p][rem * 4]), g);
        } else {
            const int c2 = c - 1152;
            const float* g = T2 + bcBase * 9 + c2 * 4;
            async_load_b128(lds_off(&sT2[c2 * 4]), g);
        }
    }
    wait_async0();
    __syncthreads();

    const int wave = tid >> 5;
    const int lane = tid & 31;
    const int h    = lane >> 4;        // lane half: K{0,1} vs K{2,3}
    const int r    = lane & 15;        // A row / B column inside 16x16 tile
    const int q    = r / 3;            // diagonal 3x3 block index (0..5)
    const int rm   = r - 3 * q;        // i (A rows) / j (B cols)
    const int qc   = (q > 4) ? 4 : q;  // clamp: block index for p=0..4 pass
    const int q2   = (q > 2) ? 2 : q;  // clamp: block index for p=5..7 pass

    float* wS = &sD[wave * 16 * SROW];
    // scatter base: lane half h owns D rows 8h..8h+7 of column r
    float* scat = wS + (8 * h) * SROW + r;
    // gather base: lane r reads D row r, cols 3q..3q+2 (contiguous -> out row i=rm)
    const float* rb = wS + r * SROW + 3 * qc;

    for (int t = 0; t < 8; ++t) {
        const int bcl = wave * 8 + t;                // each wave owns 8 bc pairs
        const float* t2 = &sT2[bcl * 9];

        // B operand: B[m][3q+j] = T2[m][j] on 5 diagonal blocks; K=3 row zeroed.
        v2f b;
        b.x = t2[6 * h + rm];                        // K0 (h=0) / K2 (h=1)
        {
            const float bmid = t2[3 + rm];
            b.y = h ? 0.f : bmid;                    // K1 (h=0) / K3 = 0 (h=1)
        }

        // A operand, pass 1: blocks q = 0..4 hold T1[p=q]
        const float* a1p = &sT1[qc][bcl * 9 + rm * 3];
        v2f a1;
        a1.x = a1p[2 * h];                           // K0 / K2
        a1.y = a1p[1];                               // K1 (h=1 slot killed by b.y=0)

        // A operand, pass 2: blocks q = 0..2 hold T1[p=5+q]
        const float* a2p = &sT1[5 + q2][bcl * 9 + rm * 3];
        v2f a2;
        a2.x = a2p[2 * h];
        a2.y = a2p[1];

        v8f c0 = {};
        v8f d1 = wmma_f32_4(a1, b, c0);              // p = 0..4
        v8f d2 = wmma_f32_4(a2, b, c0);              // p = 5..7

        // dest pointers; invalid lanes sink into sDummy (pointer select, no branch)
        float* dst1 = (r < 15) ? &sOut[qc][bcl * 9 + rm * 3] : sDummy;
        float* dst2 = (r < 9) ? &sOut[5 + q2][bcl * 9 + rm * 3] : sDummy;

        // ---- d1: scatter full D into padded scratch, gather 3 contiguous cols
        #pragma unroll
        for (int v = 0; v < 8; ++v) scat[v * SROW] = d1[v];
        dst1[0] = rb[0];
        dst1[1] = rb[1];
        dst1[2] = rb[2];

        // ---- d2: reuse scratch (in-wave DS ordering guarantees RAW/WAR safety)
        #pragma unroll
        for (int v = 0; v < 8; ++v) scat[v * SROW] = d2[v];
        dst2[0] = rb[0];
        dst2[1] = rb[1];
        dst2[2] = rb[2];
    }

    __syncthreads();

    // ---- stream the out tile back with coalesced async LDS->global stores
    for (int c = tid; c < 1152; c += THREADS) {
        const int p = c / 144, rem = c % 144;
        float* g = out + ((size_t)p * BCN + bcBase) * 9 + rem * 4;
        async_store_b128(g, lds_off(&sOut[p][rem * 4]));
    }
    wait_async0();
}

extern "C" void kernel_launch(void* const* d_in, const int* in_sizes, int n_in,
                              void* d_out, int out_size, void* d_ws, size_t ws_size,
                              hipStream_t stream) {
    const float* T1  = (const float*)d_in[0];   // (P, B, C, 3, 3)
    const float* T2  = (const float*)d_in[1];   // (B, C, 3, 3)
    float*       out = (float*)d_out;           // (P, B, C, 3, 3)

    dim3 grid(BCN / TILE);   // 16384 blocks, 256 threads (8 waves) each
    tc_bcast_wmma_kernel<<<grid, THREADS, 0, stream>>>(T1, T2, out);
}